// CrossAttention_47742856462840
// MI455X (gfx1250) — compile-verified
//
#include <hip/hip_runtime.h>
#include <hip/hip_bf16.h>
#include <math.h>

typedef __attribute__((ext_vector_type(2))) float v2f;
typedef __attribute__((ext_vector_type(8))) float v8f;

#define B_   4
#define LQ_  512
#define LK_  512
#define DQ_  256
#define DC_  256
#define H_   128

// ---------------------------------------------------------------------------
// fast device math (native gfx1250 transcendentals where available)
// ---------------------------------------------------------------------------
__device__ __forceinline__ float fast_tanh(float x) {
#if defined(__HIP_DEVICE_COMPILE__)
#if __has_builtin(__builtin_amdgcn_tanhf)
  return __builtin_amdgcn_tanhf(x);
#elif __has_builtin(__builtin_amdgcn_tanh_f32)
  return __builtin_amdgcn_tanh_f32(x);
#else
  // tanh(x) = 1 - 2/(exp(2x)+1), exp via native v_exp_f32 (exp2)
  float e = __builtin_amdgcn_exp2f(x * 2.8853900817779268f); // 2*log2(e)
  return 1.0f - 2.0f * __builtin_amdgcn_rcpf(e + 1.0f);
#endif
#else
  return tanhf(x);
#endif
}

__device__ __forceinline__ float fast_exp2(float x) {
#if defined(__HIP_DEVICE_COMPILE__)
  return __builtin_amdgcn_exp2f(x);
#else
  return exp2f(x);
#endif
}

__device__ __forceinline__ float fast_rcp(float x) {
#if defined(__HIP_DEVICE_COMPILE__)
  return __builtin_amdgcn_rcpf(x);
#else
  return 1.0f / x;
#endif
}

// ---------------------------------------------------------------------------
// Kernel 1: out[m,h] = scale * (sum_d X[m,d] * W[h,d] + bias[h])
// One 16x16 output tile per wave, V_WMMA_F32_16X16X4_F32, K-loop step 4.
// A frag (16x4): lane l holds row m = l%16, K = (l/16)*2 + {0,1}  (2 VGPRs)
// B frag (4x16): lane l holds col n = l%16, K = (l/16)*2 + {0,1}
// C/D (16x16):   vgpr r, lane l -> m = r + (l/16)*8, n = l%16
// ---------------------------------------------------------------------------
__global__ void proj_wmma(const float* __restrict__ X, const float* __restrict__ W,
                          const float* __restrict__ bias, float* __restrict__ out,
                          int M, int D, int H, float scale) {
  const int lane  = threadIdx.x & 31;
  const int wave  = threadIdx.x >> 5;
  const int mt    = blockIdx.x * 4 + wave;   // M tile index
  const int nt    = blockIdx.y;              // H tile index
  const int row   = mt * 16 + (lane & 15);
  const int col   = nt * 16 + (lane & 15);
  const int khalf = (lane >> 4) * 2;

  const float* xrow = X + (size_t)row * D + khalf;
  const float* wrow = W + (size_t)col * D + khalf;

  v8f acc = {};
#pragma unroll 8
  for (int k = 0; k < D; k += 4) {
    v2f a = *(const v2f*)(xrow + k);   // 8B aligned: row*D even, khalf even
    v2f b = *(const v2f*)(wrow + k);
    acc = __builtin_amdgcn_wmma_f32_16x16x4_f32(false, a, false, b,
                                                (short)0, acc, false, false);
  }

  const float bb = bias[col];
  const int r0 = (lane >> 4) * 8;
#pragma unroll
  for (int r = 0; r < 8; ++r) {
    int m = mt * 16 + r0 + r;
    out[(size_t)m * H + col] = scale * (acc[r] + bb);
  }
}

// ---------------------------------------------------------------------------
// Kernel 2: scores + fused softmax.
// Block = 256 threads (8 waves); wave w owns q-row q0+w; lanes parallel over k.
// cp2 staged in LDS transposed [h][k] with pad stride 65 -> conflict-free.
// Each lane accumulates 16 scores (k = lane, lane+32 per 64-chunk), then
// wave-wide shuffle max/sum softmax, writes attn row to workspace.
// ---------------------------------------------------------------------------
__global__ void scores_softmax(const float* __restrict__ qp, const float* __restrict__ cp2,
                               const float* __restrict__ Wv, const float* __restrict__ bv,
                               float* __restrict__ attn) {
  __shared__ float qps[8][H_];       // 4 KB
  __shared__ float wvs[H_];          // 0.5 KB
  __shared__ float cps[H_][65];      // 33.3 KB, padded stride (bank-conflict free)

  const int b    = blockIdx.y;
  const int q0   = blockIdx.x * 8;
  const int tid  = threadIdx.x;
  const int lane = tid & 31;
  const int wave = tid >> 5;

  for (int idx = tid; idx < 8 * H_; idx += 256)
    qps[idx >> 7][idx & (H_ - 1)] =
        qp[((size_t)b * LQ_ + q0 + (idx >> 7)) * H_ + (idx & (H_ - 1))];
  if (tid < H_) wvs[tid] = Wv[tid];
  __syncthreads();

  const float bvv = bv[0];
  const float* cpb = cp2 + (size_t)b * LK_ * H_;
  float sc[16];

  for (int c = 0; c < 8; ++c) {
    const int k0 = c * 64;
    // stage 64 k-rows, transposed: global read coalesced over h
    for (int idx = tid; idx < 64 * H_; idx += 256) {
      int kk = idx >> 7, h = idx & (H_ - 1);
      cps[h][kk] = cpb[(size_t)(k0 + kk) * H_ + h];
    }
    __syncthreads();

    float s0 = 0.f, s1 = 0.f;
#pragma unroll 4
    for (int h = 0; h < H_; ++h) {
      float t = qps[wave][h];     // LDS broadcast
      float w = wvs[h];           // LDS broadcast
      s0 += fast_tanh(t + cps[h][lane]) * w;
      s1 += fast_tanh(t + cps[h][32 + lane]) * w;
    }
    sc[c * 2]     = s0 + bvv;
    sc[c * 2 + 1] = s1 + bvv;
    __syncthreads();
  }

  // fused softmax over 512 values (16 per lane, wave32 reductions)
  float m = sc[0];
#pragma unroll
  for (int i = 1; i < 16; ++i) m = fmaxf(m, sc[i]);
#pragma unroll
  for (int off = 16; off >= 1; off >>= 1) m = fmaxf(m, __shfl_xor(m, off, 32));

  float sum = 0.f;
#pragma unroll
  for (int i = 0; i < 16; ++i) {
    sc[i] = fast_exp2((sc[i] - m) * 1.4426950408889634f);
    sum += sc[i];
  }
#pragma unroll
  for (int off = 16; off >= 1; off >>= 1) sum += __shfl_xor(sum, off, 32);
  const float inv = fast_rcp(sum);

  float* arow = attn + ((size_t)b * LQ_ + q0 + wave) * LK_;
#pragma unroll
  for (int i = 0; i < 16; ++i) {
    int k = (i >> 1) * 64 + (i & 1) * 32 + lane;
    arow[k] = sc[i] * inv;
  }
}

// ---------------------------------------------------------------------------
// Kernel 3: out[b] = attn[b] @ context[b]   (512x512 @ 512x256, f32 WMMA)
// ---------------------------------------------------------------------------
__global__ void av_wmma(const float* __restrict__ attn, const float* __restrict__ ctx,
                        float* __restrict__ out) {
  const int lane  = threadIdx.x & 31;
  const int wave  = threadIdx.x >> 5;
  const int b     = blockIdx.z;
  const int mt    = blockIdx.x * 4 + wave;  // 0..31
  const int nt    = blockIdx.y;             // 0..15
  const int row   = mt * 16 + (lane & 15);
  const int col   = nt * 16 + (lane & 15);
  const int khalf = (lane >> 4) * 2;

  const float* arow = attn + ((size_t)b * LQ_ + row) * LK_ + khalf;
  const float* cb   = ctx + (size_t)b * LK_ * DC_;

  v8f acc = {};
#pragma unroll 8
  for (int k = 0; k < LK_; k += 4) {
    v2f a = *(const v2f*)(arow + k);
    v2f bm;
    bm.x = cb[(size_t)(k + khalf) * DC_ + col];
    bm.y = cb[(size_t)(k + khalf + 1) * DC_ + col];
    acc = __builtin_amdgcn_wmma_f32_16x16x4_f32(false, a, false, bm,
                                                (short)0, acc, false, false);
  }

  const int r0 = (lane >> 4) * 8;
#pragma unroll
  for (int r = 0; r < 8; ++r) {
    int m = mt * 16 + r0 + r;
    out[((size_t)b * LQ_ + m) * DC_ + col] = acc[r];
  }
}

// ---------------------------------------------------------------------------
extern "C" void kernel_launch(void* const* d_in, const int* in_sizes, int n_in,
                              void* d_out, int out_size, void* d_ws, size_t ws_size,
                              hipStream_t stream) {
  const float* query   = (const float*)d_in[0];
  const float* context = (const float*)d_in[1];
  const float* Wq      = (const float*)d_in[2];
  const float* bq      = (const float*)d_in[3];
  const float* Wc      = (const float*)d_in[4];
  const float* bc      = (const float*)d_in[5];
  const float* Wv      = (const float*)d_in[6];
  const float* bv      = (const float*)d_in[7];
  float* out = (float*)d_out;

  float* qp   = (float*)d_ws;                       // [B*LQ, H]  1 MB
  float* cp2  = qp  + (size_t)B_ * LQ_ * H_;        // [B*LK, H]  1 MB
  float* attn = cp2 + (size_t)B_ * LK_ * H_;        // [B,LQ,LK]  4 MB

  dim3 pb(128);
  // qp' = query @ Wq^T + bq            (scale 1)
  proj_wmma<<<dim3((B_ * LQ_) / 64, H_ / 16), pb, 0, stream>>>(
      query, Wq, bq, qp, B_ * LQ_, DQ_, H_, 1.0f);
  // cp2 = 2*(context @ Wc^T + bc)      (folds the doubled context_proj)
  proj_wmma<<<dim3((B_ * LK_) / 64, H_ / 16), pb, 0, stream>>>(
      context, Wc, bc, cp2, B_ * LK_, DC_, H_, 2.0f);

  scores_softmax<<<dim3(LQ_ / 8, B_), dim3(256), 0, stream>>>(qp, cp2, Wv, bv, attn);

  av_wmma<<<dim3(LQ_ / 64, DC_ / 16, B_), pb, 0, stream>>>(attn, context, out);
}